// COFunc_12859132084469
// MI455X (gfx1250) — compile-verified
//
#include <hip/hip_runtime.h>
#include <stdint.h>

// ---------------------------------------------------------------------------
// Spring-force graph message passing:
//   dv[b,n,:] = (1/m[n]) * sum_{edges e incident to n} (+/-) k_e*(r[src]-r[dst])
//   dr[b,n,:] = v[b,n,:]
// u = (B,N,32): channels 0..15 = v, 16..31 = r.  out = (B,N,32): dv | dr.
// Memory/atomic bound (AI ~0.7 flop/byte) -> no WMMA; optimize data movement.
// ---------------------------------------------------------------------------

#define EDGES_PER_BLOCK 32
#define BLOCK_THREADS   256

#if defined(__HIP_DEVICE_COMPILE__) && \
    __has_builtin(__builtin_amdgcn_global_load_async_to_lds_b64) && \
    __has_builtin(__builtin_amdgcn_global_load_async_to_lds_b32)
#define ASYNC_LDS 1
#define AS1 __attribute__((address_space(1)))
#define AS3 __attribute__((address_space(3)))
typedef int v2i_t __attribute__((ext_vector_type(2)));
#else
#define ASYNC_LDS 0
#endif

__device__ __forceinline__ void atomic_add_f32(float* p, float v) {
#if defined(__HIP_DEVICE_COMPILE__)
    // Force native global_atomic_add_f32 (device-scope, no CAS loop).
    unsafeAtomicAdd(p, v);
#else
    atomicAdd(p, v);
#endif
}

// Pass 1: out[..., 0:16] = 0 (atomic accumulator), out[..., 16:32] = v.
// One float4 per thread, fully coalesced. total4 = B*N*8 float4 elements.
__global__ __launch_bounds__(BLOCK_THREADS)
void init_kernel(const float* __restrict__ u, float* __restrict__ out,
                 long long total4) {
    long long i = (long long)blockIdx.x * blockDim.x + threadIdx.x;
    if (i >= total4) return;
    long long node = i >> 3;     // flat (b*N + n)
    int c = (int)(i & 7);        // float4 chunk within the 32-channel row
    float4 val;
    if (c < 4) {
        val = make_float4(0.f, 0.f, 0.f, 0.f);            // dv accumulator
    } else {
        val = ((const float4*)u)[node * 8 + (c - 4)];     // dr = v
    }
    ((float4*)out)[i] = val;
}

// Pass 2: per undirected edge, compute diff = r[s]-r[d] once and scatter
// +diff*k/m[d] to node d and -diff*k/m[s] to node s, for all B batches.
// 32 edges per block; 8 threads (one per batch) per edge; edge metadata is
// staged block-wide into LDS via CDNA5 async global->LDS copies.
__global__ __launch_bounds__(BLOCK_THREADS)
void edge_scatter_kernel(const float* __restrict__ u,
                         const long long* __restrict__ ei,   // (2,E) int64
                         const float* __restrict__ ke,       // (E,)
                         const float* __restrict__ m,        // (N,)
                         float* __restrict__ out,            // (B,N,32)
                         int nE, int nN, int nB) {
    __shared__ long long s_src[EDGES_PER_BLOCK];
    __shared__ long long s_dst[EDGES_PER_BLOCK];
    __shared__ float     s_k[EDGES_PER_BLOCK];

    const int t  = threadIdx.x;
    const int e0 = blockIdx.x * EDGES_PER_BLOCK;

    if (t < EDGES_PER_BLOCK) {
        const int e = e0 + t;
        if (e < nE) {
#if ASYNC_LDS
            __builtin_amdgcn_global_load_async_to_lds_b64(
                (AS1 v2i_t*)(void*)(ei + e),
                (AS3 v2i_t*)(void*)&s_src[t], 0, 0);
            __builtin_amdgcn_global_load_async_to_lds_b64(
                (AS1 v2i_t*)(void*)(ei + nE + e),
                (AS3 v2i_t*)(void*)&s_dst[t], 0, 0);
            __builtin_amdgcn_global_load_async_to_lds_b32(
                (AS1 int*)(void*)(ke + e),
                (AS3 int*)(void*)&s_k[t], 0, 0);
#else
            s_src[t] = ei[e];
            s_dst[t] = ei[nE + e];
            s_k[t]   = ke[e];
#endif
        }
    }
#if ASYNC_LDS
    asm volatile("s_wait_asynccnt 0" ::: "memory");
#endif
    __syncthreads();

    const int le = t >> 3;                 // local edge id 0..31
    const int e  = e0 + le;
    if (e >= nE) return;

    const int   s = (int)s_src[le];
    const int   d = (int)s_dst[le];
    const float k = s_k[le];
    // Pre-scale by 1/m at accumulate time -> no finalize pass over d_out.
    const float sc_d = k / m[d];
    const float sc_s = k / m[s];

    for (int b = (t & 7); b < nB; b += 8) {
        const size_t base_s = ((size_t)b * nN + s) * 32;
        const size_t base_d = ((size_t)b * nN + d) * 32;
        const float4* rs4 = (const float4*)(u + base_s + 16);  // r[b,s,:]
        const float4* rd4 = (const float4*)(u + base_d + 16);  // r[b,d,:]
        float* od = out + base_d;                              // dv[b,d,:]
        float* os = out + base_s;                              // dv[b,s,:]
#pragma unroll
        for (int q = 0; q < 4; ++q) {
            const float4 a = rs4[q];
            const float4 c = rd4[q];
            const float dx = a.x - c.x;
            const float dy = a.y - c.y;
            const float dz = a.z - c.z;
            const float dw = a.w - c.w;
            atomic_add_f32(od + 4 * q + 0,  dx * sc_d);
            atomic_add_f32(od + 4 * q + 1,  dy * sc_d);
            atomic_add_f32(od + 4 * q + 2,  dz * sc_d);
            atomic_add_f32(od + 4 * q + 3,  dw * sc_d);
            atomic_add_f32(os + 4 * q + 0, -dx * sc_s);
            atomic_add_f32(os + 4 * q + 1, -dy * sc_s);
            atomic_add_f32(os + 4 * q + 2, -dz * sc_s);
            atomic_add_f32(os + 4 * q + 3, -dw * sc_s);
        }
    }
}

extern "C" void kernel_launch(void* const* d_in, const int* in_sizes, int n_in,
                              void* d_out, int out_size, void* d_ws, size_t ws_size,
                              hipStream_t stream) {
    // setup_inputs order: t(1), u(B*N*32), edge_index(2E) int64, k_e(E), m(N)
    const float*     u  = (const float*)d_in[1];
    const long long* ei = (const long long*)d_in[2];
    const float*     ke = (const float*)d_in[3];
    const float*     m  = (const float*)d_in[4];
    float*           out = (float*)d_out;

    const int nE = in_sizes[3];
    const int nN = in_sizes[4];
    const int nB = in_sizes[1] / (nN * 32);

    const long long total4 = (long long)nB * nN * 8;
    const int init_blocks = (int)((total4 + BLOCK_THREADS - 1) / BLOCK_THREADS);
    init_kernel<<<init_blocks, BLOCK_THREADS, 0, stream>>>(u, out, total4);

    const int edge_blocks = (nE + EDGES_PER_BLOCK - 1) / EDGES_PER_BLOCK;
    edge_scatter_kernel<<<edge_blocks, BLOCK_THREADS, 0, stream>>>(
        u, ei, ke, m, out, nE, nN, nB);
}